// IMM_Particle_Filter_42391327211992
// MI455X (gfx1250) — compile-verified
//
#include <hip/hip_runtime.h>
#include <hip/hip_bf16.h>
#include <math.h>

#define B_   16
#define N_   8192
#define NK_  8
#define D_   64
#define OBS_ 32
#define PPM_ 1024

typedef __attribute__((ext_vector_type(2))) float v2f;
typedef __attribute__((ext_vector_type(8))) float v8f;

// ---------------------------------------------------------------------------
// Kernel 1: adj = log_softmax_k(logP[n/PPM,k] + x.G) + tw ; store e = exp(adj)
// in transposed layout e_t[(b*NK+k)*N + n] for the per-(b,k) scans.
// ---------------------------------------------------------------------------
__global__ __launch_bounds__(256) void k1_regime(
    const float* __restrict__ x_t, const float* __restrict__ tw,
    const float* __restrict__ logP, const float* __restrict__ G,
    float* __restrict__ e_t) {
  __shared__ float Gs[D_ * NK_];
  __shared__ float Ps[NK_ * NK_];
  int tid = threadIdx.x;
  for (int i = tid; i < D_ * NK_; i += 256) Gs[i] = G[i];
  if (tid < NK_ * NK_) Ps[tid] = logP[tid];
  __syncthreads();

  int g = blockIdx.x * 256 + tid;          // (b*N + n)
  int n = g % N_;
  int b = g / N_;
  int mid = n / PPM_;
  const float* x = x_t + (size_t)g * D_;

  float l[NK_];
#pragma unroll
  for (int k = 0; k < NK_; ++k) l[k] = Ps[mid * NK_ + k];
  for (int d = 0; d < D_; ++d) {
    float xv = x[d];
#pragma unroll
    for (int k = 0; k < NK_; ++k) l[k] = fmaf(xv, Gs[d * NK_ + k], l[k]);
  }
  float mx = l[0];
#pragma unroll
  for (int k = 1; k < NK_; ++k) mx = fmaxf(mx, l[k]);
  float se = 0.f, ek[NK_];
#pragma unroll
  for (int k = 0; k < NK_; ++k) { ek[k] = __expf(l[k] - mx); se += ek[k]; }
  float scale = __expf(tw[g]) / se;        // exp(adj) = exp(l-mx)*exp(tw)/se
#pragma unroll
  for (int k = 0; k < NK_; ++k)
    e_t[((size_t)(b * NK_ + k)) * N_ + n] = ek[k] * scale;
}

// ---------------------------------------------------------------------------
// Kernel 2: in-place inclusive prefix sum over N=8192 per (b,k); store totals.
// One workgroup (256 thr) per scan; each thread owns 32 contiguous elements.
// ---------------------------------------------------------------------------
__global__ __launch_bounds__(256) void k2_scan(float* __restrict__ cum,
                                               float* __restrict__ totals) {
  __shared__ float sd[256];
  int bk = blockIdx.x, tid = threadIdx.x;
  float* row = cum + (size_t)bk * N_;
  int base = tid * 32;
  float v[32];
  float s = 0.f;
#pragma unroll
  for (int i = 0; i < 32; ++i) { s += row[base + i]; v[i] = s; }
  sd[tid] = s;
  __syncthreads();
  for (int off = 1; off < 256; off <<= 1) {
    float t = (tid >= off) ? sd[tid - off] : 0.f;
    __syncthreads();
    sd[tid] += t;
    __syncthreads();
  }
  float excl = sd[tid] - s;
#pragma unroll
  for (int i = 0; i < 32; ++i) row[base + i] = v[i] + excl;
  if (tid == 0) totals[bk] = sd[255];
}

// ---------------------------------------------------------------------------
// Kernel 3: systematic resampling; lower_bound on unnormalized cum vs u*total.
// ---------------------------------------------------------------------------
__global__ __launch_bounds__(256) void k3_resample(
    const float* __restrict__ cum, const float* __restrict__ totals,
    const float* __restrict__ u0, int* __restrict__ idx) {
  int g = blockIdx.x * 256 + threadIdx.x;
  int b = g / N_, n = g % N_;
  int k = n / PPM_, p = n % PPM_;
  float u = ((float)p + u0[b * NK_ + k]) * (1.0f / (float)PPM_);
  float v = u * totals[b * NK_ + k];
  const float* c = cum + (size_t)(b * NK_ + k) * N_;
  int lo = 0, hi = N_;
  while (lo < hi) {
    int m = (lo + hi) >> 1;
    if (c[m] < v) lo = m + 1; else hi = m;
  }
  if (lo > N_ - 1) lo = N_ - 1;
  idx[g] = lo;
}

// ---------------------------------------------------------------------------
// Kernel 4: gather + x_new = xs.A^T + b + 0.1*noise (WMMA f32 16x16x4),
// pred = x_new.H^T (WMMA), residual norm -> logw.
// One wave = one 16-row tile (tile never crosses a model: 16 | PPM).
// Per-wave LDS tile, row stride 68 floats (bank-conflict padding).
// ---------------------------------------------------------------------------
__global__ __launch_bounds__(128) void k4_main(
    const float* __restrict__ x_t, const float* __restrict__ y,
    const float* __restrict__ A, const float* __restrict__ bvec,
    const float* __restrict__ H, const float* __restrict__ noise,
    const float* __restrict__ totals, const int* __restrict__ idx,
    float* __restrict__ out_x, float* __restrict__ logw) {
  __shared__ float lds[4 * 16 * 68];
  int lane = threadIdx.x & 31;
  int wv   = threadIdx.x >> 5;
  float* tb = lds + wv * 16 * 68;

  int tile = blockIdx.x * 4 + wv;
  int row0 = tile * 16;                    // flat (b*N + n) row base
  int bI   = row0 / N_;
  int n0   = row0 % N_;
  int k    = n0 / PPM_;

  int h  = lane >> 4;                      // lane half (K split for WMMA frags)
  int lc = lane & 15;

  // gather resampled source rows into private LDS tile (16 x 64)
  int gv = idx[row0 + lc];
#pragma unroll
  for (int i = 0; i < 16; ++i) {
    int ridx = __shfl(gv, i, 32);
    const float2 xv =
        *(const float2*)(x_t + ((size_t)bI * N_ + ridx) * D_ + lane * 2);
    tb[i * 68 + lane * 2]     = xv.x;
    tb[i * 68 + lane * 2 + 1] = xv.y;
  }

  // GEMM1: x_new[p,e] = sum_d xs[p,d] * A[k,e,d]
  v8f acc[4] = {};
  const float* Ak = A + (size_t)k * D_ * D_;
#pragma unroll
  for (int ct = 0; ct < 4; ++ct) {
    int col = ct * 16 + lc;
#pragma unroll
    for (int kb = 0; kb < D_; kb += 4) {
      int kk = kb + 2 * h;
      v2f a; a.x = tb[lc * 68 + kk]; a.y = tb[lc * 68 + kk + 1];
      const float2 bf = *(const float2*)(Ak + (size_t)col * D_ + kk);
      v2f bb; bb.x = bf.x; bb.y = bf.y;
      acc[ct] = __builtin_amdgcn_wmma_f32_16x16x4_f32(
          false, a, false, bb, (short)0, acc[ct], false, false);
    }
  }

  // epilogue: + b[k,e] + 0.1*noise ; store to d_out and back into LDS tile
  const float* bk_ = bvec + k * D_;
#pragma unroll
  for (int ct = 0; ct < 4; ++ct) {
    int col = ct * 16 + lc;
    float bcol = bk_[col];
#pragma unroll
    for (int j = 0; j < 8; ++j) {
      int row = j + 8 * h;
      size_t gRow = (size_t)(row0 + row);
      float val = acc[ct][j] + bcol + 0.1f * noise[gRow * D_ + col];
      out_x[gRow * D_ + col] = val;
      tb[row * 68 + col] = val;            // wave-ordered DS: safe, no barrier
    }
  }

  // GEMM2: pred[p,o] = sum_d x_new[p,d] * H[k,o,d]
  v8f pacc[2] = {};
  const float* Hk = H + (size_t)k * OBS_ * D_;
#pragma unroll
  for (int ot = 0; ot < 2; ++ot) {
    int o = ot * 16 + lc;
#pragma unroll
    for (int kb = 0; kb < D_; kb += 4) {
      int kk = kb + 2 * h;
      v2f a; a.x = tb[lc * 68 + kk]; a.y = tb[lc * 68 + kk + 1];
      const float2 hf = *(const float2*)(Hk + (size_t)o * D_ + kk);
      v2f bb; bb.x = hf.x; bb.y = hf.y;
      pacc[ot] = __builtin_amdgcn_wmma_f32_16x16x4_f32(
          false, a, false, bb, (short)0, pacc[ot], false, false);
    }
  }

  // residual -> logf -> logw  (logw = logf - log(PPM) + tot[b,k])
  float y0 = y[bI * OBS_ + lc];
  float y1 = y[bI * OBS_ + 16 + lc];
  float tot = __logf(totals[bI * NK_ + k]);
#pragma unroll
  for (int j = 0; j < 8; ++j) {
    float r0 = y0 - pacc[0][j];
    float r1 = y1 - pacc[1][j];
    float s = r0 * r0 + r1 * r1;
#pragma unroll
    for (int m = 1; m < 16; m <<= 1) s += __shfl_xor(s, m, 32);
    // s now = full 32-obs squared residual for row (j + 8*h)
    float lf = -s - 18.3156782f;           // -0.5*s/0.5 - 0.5*OBS*log(2*pi*0.5)
    float lw = lf - 6.93147181f + tot;     // - log(PPM)
    if (lc == 0) logw[row0 + j + 8 * h] = lw;
  }
}

// ---------------------------------------------------------------------------
// Kernel 5: per-batch logsumexp over N of logw.
// ---------------------------------------------------------------------------
__global__ __launch_bounds__(256) void k5_lse(const float* __restrict__ logw,
                                              float* __restrict__ lse) {
  __shared__ float sd[256];
  int b = blockIdx.x, tid = threadIdx.x;
  const float* r = logw + (size_t)b * N_;
  float m = -3.0e38f;
  for (int i = tid; i < N_; i += 256) m = fmaxf(m, r[i]);
  sd[tid] = m; __syncthreads();
  for (int off = 128; off > 0; off >>= 1) {
    if (tid < off) sd[tid] = fmaxf(sd[tid], sd[tid + off]);
    __syncthreads();
  }
  float mx = sd[0]; __syncthreads();
  float s = 0.f;
  for (int i = tid; i < N_; i += 256) s += __expf(r[i] - mx);
  sd[tid] = s; __syncthreads();
  for (int off = 128; off > 0; off >>= 1) {
    if (tid < off) sd[tid] += sd[tid + off];
    __syncthreads();
  }
  if (tid == 0) lse[b] = mx + __logf(sd[0]);
}

// ---------------------------------------------------------------------------
// Kernel 6: log_norm = logw - lse[b]
// ---------------------------------------------------------------------------
__global__ __launch_bounds__(256) void k6_norm(const float* __restrict__ logw,
                                               const float* __restrict__ lse,
                                               float* __restrict__ outl) {
  int g = blockIdx.x * 256 + threadIdx.x;
  outl[g] = logw[g] - lse[g / N_];
}

// ---------------------------------------------------------------------------
extern "C" void kernel_launch(void* const* d_in, const int* in_sizes, int n_in,
                              void* d_out, int out_size, void* d_ws,
                              size_t ws_size, hipStream_t stream) {
  const float* x_t   = (const float*)d_in[0];
  const float* tw    = (const float*)d_in[1];
  const float* y     = (const float*)d_in[2];
  const float* A     = (const float*)d_in[3];
  const float* bvec  = (const float*)d_in[4];
  const float* H     = (const float*)d_in[5];
  const float* logP  = (const float*)d_in[6];
  const float* G     = (const float*)d_in[7];
  const float* noise = (const float*)d_in[8];
  const float* u0    = (const float*)d_in[9];

  float* ws = (float*)d_ws;
  float* cum    = ws;                           // B*NK*N   = 1,048,576 f
  float* totals = ws + 1048576;                 // B*NK     = 128 f
  float* logw   = ws + 1048704;                 // B*N      = 131,072 f
  float* lse    = ws + 1179776;                 // B        = 16 f
  int*   idxb   = (int*)(ws + 1179792);         // B*N ints

  float* out_x = (float*)d_out;                 // (B,N,D)
  float* out_l = out_x + (size_t)B_ * N_ * D_;  // (B,N)

  k1_regime <<<(B_ * N_) / 256, 256, 0, stream>>>(x_t, tw, logP, G, cum);
  k2_scan   <<<B_ * NK_,        256, 0, stream>>>(cum, totals);
  k3_resample<<<(B_ * N_) / 256, 256, 0, stream>>>(cum, totals, u0, idxb);
  k4_main   <<<(B_ * N_) / (16 * 4), 128, 0, stream>>>(
      x_t, y, A, bvec, H, noise, totals, idxb, out_x, logw);
  k5_lse    <<<B_,              256, 0, stream>>>(logw, lse);
  k6_norm   <<<(B_ * N_) / 256, 256, 0, stream>>>(logw, lse, out_l);
}